// SupervisedGraphSage_47914655154262
// MI455X (gfx1250) — compile-verified
//
#include <hip/hip_runtime.h>
#include <math.h>

typedef __attribute__((ext_vector_type(2))) float v2f;
typedef __attribute__((ext_vector_type(8))) float v8f;

#define N_NODES 65536
#define DEG     16
#define D       64
#define D2      128
#define NB1     1024
#define NNEG    10
#define GTOT    (NB1 + NB1 + NNEG)   // 2058 output rows
#define EPSN    1e-12f
#define XPAD    132                  // padded LDS row stride (floats): 132*4 % 16 == 0

// ---------------------------------------------------------------------------
// Hop 1: for every node, X = [feat | sum(feat[neighbors])] (16x128 tile in LDS),
// h1 = tanh(X @ W1^T + b1) via V_WMMA_F32_16X16X4_F32, stored to workspace.
// One wave (32 threads) per 16-node tile.
// ---------------------------------------------------------------------------
__global__ __launch_bounds__(32) void graphsage_hop1(
    const float* __restrict__ feat, const int* __restrict__ adj,
    const float* __restrict__ W1,   const float* __restrict__ b1,
    float* __restrict__ h1)
{
    __shared__ float Xs[16][XPAD];

    const int lane = threadIdx.x;
    const int m    = lane & 15;       // row within tile
    const int half = lane >> 4;       // which 32-col half this lane owns
    const int tile = blockIdx.x;
    const int node = tile * 16 + m;   // N is a multiple of 16

    // ---- cooperative gather: self features + neighbor sum (float4 vectorized)
    {
        const float4* selfp = (const float4*)(feat + (size_t)node * D + half * 32);
        float4* xlo = (float4*)&Xs[m][half * 32];        // cols [0,64)   : self
        float4* xhi = (float4*)&Xs[m][D + half * 32];    // cols [64,128) : neighbor sum
        #pragma unroll
        for (int i = 0; i < 8; ++i) xlo[i] = selfp[i];

        float4 acc[8];
        #pragma unroll
        for (int i = 0; i < 8; ++i) acc[i] = make_float4(0.f, 0.f, 0.f, 0.f);
        for (int j = 0; j < DEG; ++j) {
            const int nb = adj[(size_t)node * DEG + j];
            const float4* p = (const float4*)(feat + (size_t)nb * D + half * 32);
            #pragma unroll
            for (int i = 0; i < 8; ++i) {
                float4 v = p[i];
                acc[i].x += v.x; acc[i].y += v.y; acc[i].z += v.z; acc[i].w += v.w;
            }
        }
        #pragma unroll
        for (int i = 0; i < 8; ++i) xhi[i] = acc[i];
    }
    __syncthreads();

    // ---- GEMM: out[16,64] = X[16,128] @ W1^T[128,64], K tiled by 4
    const int kb = half * 2;          // K-pair offset per WMMA f32 A-layout
    v8f zero = {};
    v8f c[4];
    #pragma unroll
    for (int nt = 0; nt < 4; ++nt) c[nt] = zero;

    for (int k0 = 0; k0 < 2 * D; k0 += 4) {
        v2f a = *(const v2f*)&Xs[m][k0 + kb];
        #pragma unroll
        for (int nt = 0; nt < 4; ++nt) {
            const int col = nt * 16 + m;            // B lane index n == lane&15
            v2f b = *(const v2f*)&W1[(size_t)col * (2 * D) + k0 + kb];
            c[nt] = __builtin_amdgcn_wmma_f32_16x16x4_f32(
                false, a, false, b, (short)0, c[nt], false, false);
        }
    }

    // ---- epilogue: bias + tanh, scatter to h1 workspace
    #pragma unroll
    for (int nt = 0; nt < 4; ++nt) {
        const int   col  = nt * 16 + m;
        const float bias = b1[col];
        #pragma unroll
        for (int r = 0; r < 8; ++r) {
            const int row = r + half * 8;
            h1[(size_t)(tile * 16 + row) * D + col] = tanhf(c[nt][r] + bias);
        }
    }
}

// ---------------------------------------------------------------------------
// Hop 2: for the 2058 query rows (inputs1 | inputs2 | neg),
// X = [feat[node] | sum(h1[neighbors])], out = normalize(X @ W2^T + b2).
// One wave per 16-row tile; last tile masked at store time.
// ---------------------------------------------------------------------------
__global__ __launch_bounds__(32) void graphsage_hop2(
    const float* __restrict__ feat, const int* __restrict__ adj,
    const int* __restrict__ in1, const int* __restrict__ in2,
    const int* __restrict__ ineg,
    const float* __restrict__ W2, const float* __restrict__ b2,
    const float* __restrict__ h1, float* __restrict__ out)
{
    __shared__ float Xs[16][XPAD];

    const int lane = threadIdx.x;
    const int m    = lane & 15;
    const int half = lane >> 4;
    const int tile = blockIdx.x;
    const int g    = tile * 16 + m;

    int node;
    if      (g < NB1)      node = in1[g];
    else if (g < 2 * NB1)  node = in2[g - NB1];
    else if (g < GTOT)     node = ineg[g - 2 * NB1];
    else                   node = 0;              // dummy row, masked on store

    // ---- gather: self raw features + neighbor h1 sum
    {
        const float4* selfp = (const float4*)(feat + (size_t)node * D + half * 32);
        float4* xlo = (float4*)&Xs[m][half * 32];
        float4* xhi = (float4*)&Xs[m][D + half * 32];
        #pragma unroll
        for (int i = 0; i < 8; ++i) xlo[i] = selfp[i];

        float4 acc[8];
        #pragma unroll
        for (int i = 0; i < 8; ++i) acc[i] = make_float4(0.f, 0.f, 0.f, 0.f);
        for (int j = 0; j < DEG; ++j) {
            const int nb = adj[(size_t)node * DEG + j];
            const float4* p = (const float4*)(h1 + (size_t)nb * D + half * 32);
            #pragma unroll
            for (int i = 0; i < 8; ++i) {
                float4 v = p[i];
                acc[i].x += v.x; acc[i].y += v.y; acc[i].z += v.z; acc[i].w += v.w;
            }
        }
        #pragma unroll
        for (int i = 0; i < 8; ++i) xhi[i] = acc[i];
    }
    __syncthreads();

    // ---- GEMM: out[16,128] = X[16,128] @ W2^T[128,128]
    const int kb = half * 2;
    v8f zero = {};
    v8f c[8];
    #pragma unroll
    for (int nt = 0; nt < 8; ++nt) c[nt] = zero;

    for (int k0 = 0; k0 < D2; k0 += 4) {
        v2f a = *(const v2f*)&Xs[m][k0 + kb];
        #pragma unroll
        for (int nt = 0; nt < 8; ++nt) {
            const int col = nt * 16 + m;
            v2f b = *(const v2f*)&W2[(size_t)col * D2 + k0 + kb];
            c[nt] = __builtin_amdgcn_wmma_f32_16x16x4_f32(
                false, a, false, b, (short)0, c[nt], false, false);
        }
    }

    // ---- epilogue: bias, row L2-norm (cross-lane within 16-lane half), store
    float bias[8];
    #pragma unroll
    for (int nt = 0; nt < 8; ++nt) bias[nt] = b2[nt * 16 + m];

    #pragma unroll
    for (int r = 0; r < 8; ++r) {
        float vals[8];
        float ss = 0.f;
        #pragma unroll
        for (int nt = 0; nt < 8; ++nt) {
            const float v = c[nt][r] + bias[nt];
            vals[nt] = v;
            ss += v * v;
        }
        // reduce across the 16 lanes holding this row (xor<16 stays in the half)
        ss += __shfl_xor(ss, 1);
        ss += __shfl_xor(ss, 2);
        ss += __shfl_xor(ss, 4);
        ss += __shfl_xor(ss, 8);
        const float inv = 1.0f / fmaxf(sqrtf(ss), EPSN);

        const int grow = tile * 16 + r + half * 8;
        if (grow < GTOT) {
            #pragma unroll
            for (int nt = 0; nt < 8; ++nt)
                out[(size_t)grow * D2 + nt * 16 + m] = vals[nt] * inv;
        }
    }
}

// ---------------------------------------------------------------------------
extern "C" void kernel_launch(void* const* d_in, const int* in_sizes, int n_in,
                              void* d_out, int out_size, void* d_ws, size_t ws_size,
                              hipStream_t stream) {
    const float* feat = (const float*)d_in[0];
    const int*   adj  = (const int*)  d_in[1];
    const int*   in1  = (const int*)  d_in[2];
    const int*   in2  = (const int*)  d_in[3];
    const int*   ineg = (const int*)  d_in[4];
    const float* W1   = (const float*)d_in[5];
    const float* b1   = (const float*)d_in[6];
    const float* W2   = (const float*)d_in[7];
    const float* b2   = (const float*)d_in[8];
    float*       outp = (float*)d_out;
    float*       h1   = (float*)d_ws;          // N * D * 4 = 16 MB scratch

    graphsage_hop1<<<N_NODES / 16, 32, 0, stream>>>(feat, adj, W1, b1, h1);
    graphsage_hop2<<<(GTOT + 15) / 16, 32, 0, stream>>>(feat, adj, in1, in2, ineg,
                                                        W2, b2, h1, outp);
}